// Decoder_13194139533464
// MI455X (gfx1250) — compile-verified
//
#include <hip/hip_runtime.h>

typedef float v2f __attribute__((ext_vector_type(2)));
typedef float v8f __attribute__((ext_vector_type(8)));

#define S_LEN 4096
#define B_N   4
#define L_DIM 256

// d_ws float layout
#define WS_G   0      // 9 floats: g[i*3+j], i = q-feature (p,x,1), j = k-feature
#define WS_C2  16     // 128: Wv @ a2_w
#define WS_D2  160    // 128: bv @ a2_w + a2_b + a3_b
#define WS_E2  288    // 128: a3_w[0]
#define WS_M   1024   // B*S: scalar attention moment m[b,s]

// ---------------------------------------------------------------------------
// Kernel 1: tiny precompute. G = 3x3 Gram matrix of {Wq0,Wq1,bq}x{Wk0,Wk1,bk}
// (dots over L=256), plus the collapsed tail vectors c2/d2/e2 (128 each).
// ---------------------------------------------------------------------------
__global__ __launch_bounds__(256)
void precompute_kernel(const float* __restrict__ wq_w, const float* __restrict__ wq_b,
                       const float* __restrict__ wk_w, const float* __restrict__ wk_b,
                       const float* __restrict__ wv_w, const float* __restrict__ wv_b,
                       const float* __restrict__ a2_w, const float* __restrict__ a2_b,
                       const float* __restrict__ a3_w, const float* __restrict__ a3_b,
                       float* __restrict__ ws) {
  __shared__ float red[256];
  const int tid = threadIdx.x;
  float qv[3], kv[3];
  qv[0] = wq_w[tid]; qv[1] = wq_w[L_DIM + tid]; qv[2] = wq_b[tid];
  kv[0] = wk_w[tid]; kv[1] = wk_w[L_DIM + tid]; kv[2] = wk_b[tid];
  for (int i = 0; i < 3; ++i) {
    for (int j = 0; j < 3; ++j) {
      red[tid] = qv[i] * kv[j];
      __syncthreads();
      for (int off = 128; off; off >>= 1) {
        if (tid < off) red[tid] += red[tid + off];
        __syncthreads();
      }
      if (tid == 0) ws[WS_G + i * 3 + j] = red[0];
      __syncthreads();
    }
  }
  if (tid < 128) {
    float c2 = 0.f, d2 = 0.f;
    for (int l = 0; l < L_DIM; ++l) {
      float w = a2_w[l * 128 + tid];
      c2 = fmaf(wv_w[l], w, c2);
      d2 = fmaf(wv_b[l], w, d2);
    }
    ws[WS_C2 + tid] = c2;
    ws[WS_D2 + tid] = d2 + a2_b[tid] + a3_b[tid];
    ws[WS_E2 + tid] = a3_w[tid];
  }
}

// ---------------------------------------------------------------------------
// Kernel 2: causal softmax via WMMA f32 16x16x4, fully branchless inner loop.
// Each wave owns 16 output rows s. One v_wmma per 16x16 t-tile computes the
// transposed score tile D[t,s] = cp_t*(alpha_s*p_t + beta_s*x_t + gamma_s):
//   A (16x4, M=t): lanes 0-15 K0/K1 = cp_t*p_t, cp_t*x_t (M=lane);
//                  lanes 16-31 K2/K3 = cp_t, 0           (M=lane-16)
//   B (4x16, N=s): VGPR0 = {alpha | gamma}, VGPR1 = {beta | 0} per lane half.
// LDS holds a_s4[t] = (cp*p, cp*x, cp, 0) -> single ds_load_b128 per tile,
// A-halves built arithmetically (no divergent loads). Diagonal tile is peeled
// with a precomputed per-lane additive mask. Column s lives in lane pair
// {n, n+16} x 8 VGPRs: reductions = 7 in-lane ops + one shfl_xor(16).
// ---------------------------------------------------------------------------
#define ROWS_PER_WG 64
__global__ __launch_bounds__(128)
void attn_kernel(const float* __restrict__ tar_position,
                 const float* __restrict__ current_pos,
                 const float* __restrict__ tar_inp,
                 const float* __restrict__ ws_g,
                 float* __restrict__ ws_m) {
  extern __shared__ char smem_raw[];
  float4* a_s4 = (float4*)smem_raw;                      // 4096 * 16B = 64 KB
  float*  x_s  = (float*)(smem_raw + S_LEN * sizeof(float4)); // 16 KB

  const int wgPerBatch = S_LEN / ROWS_PER_WG;            // 64
  const int b     = blockIdx.x / wgPerBatch;
  const int sbase = (blockIdx.x % wgPerBatch) * ROWS_PER_WG;
  const int nload = sbase + ROWS_PER_WG;                 // only [0, sbase+64) needed

  const float* cp_g = current_pos + b * S_LEN;
  const float* p_g  = tar_position + b * S_LEN;
  const float* x_g  = tar_inp + b * S_LEN;
  for (int i = threadIdx.x; i < nload; i += blockDim.x) {
    const float cp = cp_g[i], p = p_g[i], x = x_g[i];
    a_s4[i] = make_float4(cp * p, cp * x, cp, 0.f);
    x_s[i]  = x;
  }
  __syncthreads();

  const float g00 = ws_g[0], g01 = ws_g[1], g02 = ws_g[2];
  const float g10 = ws_g[3], g11 = ws_g[4], g12 = ws_g[5];
  const float g20 = ws_g[6], g21 = ws_g[7], g22 = ws_g[8];

  const int wave = threadIdx.x >> 5;
  const int lane = threadIdx.x & 31;
  const int n    = lane & 15;
  const int half = lane >> 4;
  const int s0   = sbase + wave * 16;
  const int s    = s0 + n;

  // Per-row coefficients: alpha/beta/gamma already folded with cp_s and scale.
  const float4 avs  = a_s4[s];                           // (cp*p, cp*x, cp, 0)
  const float scale = 0.0625f;                           // 1/sqrt(256)
  const float alpha = scale * (g00 * avs.x + g10 * avs.y + g20 * avs.z);
  const float beta  = scale * (g01 * avs.x + g11 * avs.y + g21 * avs.z);
  const float gamma = scale * (g02 * avs.x + g12 * avs.y + g22 * avs.z);

  v2f bmat;                                              // loop-invariant B tile
  bmat.x = half ? gamma : alpha;                         // K=0 | K=2
  bmat.y = half ? 0.0f  : beta;                          // K=1 | K=3

  const float hf   = (float)half;                        // branchless A-half select
  const float omhf = 1.0f - hf;

  // Loop-invariant diagonal mask: lane's acc[i] holds t = t0 + half*8 + i,
  // masked iff (half*8 + i) > n on the diagonal tile.
  float msel[8];
#pragma unroll
  for (int i = 0; i < 8; ++i)
    msel[i] = ((half << 3) + i > n) ? -3.0e38f : 0.0f;

  float run_m = -3.0e38f, run_s = 0.0f, run_w = 0.0f;

  auto process_tile = [&](int t0, bool diag) {
    const int t = t0 + n;                                // same t for both halves
    const float4 av = a_s4[t];                           // ds_load_b128
    v2f amat;
    amat.x = fmaf(omhf, av.x, hf * av.z);                // half? cp : cp*p
    amat.y = omhf * av.y;                                // half? 0  : cp*x

    v8f acc = {};
    acc = __builtin_amdgcn_wmma_f32_16x16x4_f32(false, amat, false, bmat,
                                                (short)0, acc, false, false);
    if (diag) {
#pragma unroll
      for (int i = 0; i < 8; ++i) acc[i] += msel[i];     // exp() -> exact 0
    }

    const int tb = t0 + (half << 3);                     // 8-aligned
    const float4* x4 = (const float4*)x_s;
    const float4 xlo = x4[tb >> 2];                      // ds_load_b128 x2
    const float4 xhi = x4[(tb >> 2) + 1];
    const float xv[8] = {xlo.x, xlo.y, xlo.z, xlo.w, xhi.x, xhi.y, xhi.z, xhi.w};

    float tm = acc[0];
#pragma unroll
    for (int i = 1; i < 8; ++i) tm = fmaxf(tm, acc[i]);
    tm = fmaxf(tm, __shfl_xor(tm, 16, 32));              // join lane halves

    const float nm   = fmaxf(run_m, tm);
    const float corr = __expf(run_m - nm);
    float ssum = 0.f, wsum = 0.f;
#pragma unroll
    for (int i = 0; i < 8; ++i) {
      const float e = __expf(acc[i] - nm);
      ssum += e;
      wsum = fmaf(e, xv[i], wsum);
    }
    ssum += __shfl_xor(ssum, 16, 32);
    wsum += __shfl_xor(wsum, 16, 32);
    run_s = run_s * corr + ssum;
    run_w = run_w * corr + wsum;
    run_m = nm;
  };

#pragma unroll 2
  for (int t0 = 0; t0 < s0; t0 += 16) process_tile(t0, false);
  process_tile(s0, true);                                // peeled diagonal tile

  if (half == 0) ws_m[b * S_LEN + s] = run_w / run_s;
}

// ---------------------------------------------------------------------------
// Kernel 3: collapsed MLP tail. h = leaky(m*c2 + cp*e2 + d2) (128),
// z = leaky(h @ a4 + b4) (32), out = z @ a5 + b5 (2). a4 cached in LDS
// (uniform reads -> conflict-free broadcast). One thread per (b,s).
// ---------------------------------------------------------------------------
__global__ __launch_bounds__(256)
void tail_kernel(const float* __restrict__ current_pos,
                 const float* __restrict__ ws,
                 const float* __restrict__ a4_w, const float* __restrict__ a4_b,
                 const float* __restrict__ a5_w, const float* __restrict__ a5_b,
                 float* __restrict__ out) {
  __shared__ float a4s[128 * 32];
  __shared__ float c2s[128], d2s[128], e2s[128];
  __shared__ float a4bs[32], a5ws[64], a5bs[2];
  const int tid = threadIdx.x;
  for (int i = tid; i < 128 * 32; i += 256) a4s[i] = a4_w[i];
  if (tid < 128) { c2s[tid] = ws[WS_C2 + tid]; d2s[tid] = ws[WS_D2 + tid]; e2s[tid] = ws[WS_E2 + tid]; }
  if (tid < 64)  a5ws[tid] = a5_w[tid];
  if (tid < 32)  a4bs[tid] = a4_b[tid];
  if (tid < 2)   a5bs[tid] = a5_b[tid];
  __syncthreads();

  const int idx = blockIdx.x * 256 + tid;                // 0 .. B*S-1
  const float m  = ws[WS_M + idx];
  const float cp = current_pos[idx];

  float z[32];
#pragma unroll
  for (int o = 0; o < 32; ++o) z[o] = a4bs[o];

  for (int j = 0; j < 128; ++j) {
    float h = fmaf(m, c2s[j], fmaf(cp, e2s[j], d2s[j]));
    h = (h > 0.f) ? h : 0.2f * h;
#pragma unroll
    for (int o = 0; o < 32; ++o) z[o] = fmaf(h, a4s[j * 32 + o], z[o]);
  }

  float y0 = a5bs[0], y1 = a5bs[1];
#pragma unroll
  for (int o = 0; o < 32; ++o) {
    const float zz = (z[o] > 0.f) ? z[o] : 0.2f * z[o];
    y0 = fmaf(zz, a5ws[o * 2 + 0], y0);
    y1 = fmaf(zz, a5ws[o * 2 + 1], y1);
  }
  out[idx * 2 + 0] = y0;
  out[idx * 2 + 1] = y1;
}

// ---------------------------------------------------------------------------
extern "C" void kernel_launch(void* const* d_in, const int* in_sizes, int n_in,
                              void* d_out, int out_size, void* d_ws, size_t ws_size,
                              hipStream_t stream) {
  const float* tar_position = (const float*)d_in[0];
  const float* current_pos  = (const float*)d_in[1];
  const float* tar_inp      = (const float*)d_in[2];
  // d_in[3] training, d_in[4] pos_mask (unused), d_in[5] tar_mask (known causal)
  const float* wq_w = (const float*)d_in[6];
  const float* wq_b = (const float*)d_in[7];
  const float* wk_w = (const float*)d_in[8];
  const float* wk_b = (const float*)d_in[9];
  const float* wv_w = (const float*)d_in[10];
  const float* wv_b = (const float*)d_in[11];
  const float* a2_w = (const float*)d_in[12];
  const float* a2_b = (const float*)d_in[13];
  const float* a3_w = (const float*)d_in[14];
  const float* a3_b = (const float*)d_in[15];
  const float* a4_w = (const float*)d_in[16];
  const float* a4_b = (const float*)d_in[17];
  const float* a5_w = (const float*)d_in[18];
  const float* a5_b = (const float*)d_in[19];

  float* ws   = (float*)d_ws;
  float* outp = (float*)d_out;

  precompute_kernel<<<1, 256, 0, stream>>>(wq_w, wq_b, wk_w, wk_b, wv_w, wv_b,
                                           a2_w, a2_b, a3_w, a3_b, ws);

  const int nblk = B_N * (S_LEN / ROWS_PER_WG);          // 256 workgroups, 4 waves each
  const size_t shmem = S_LEN * sizeof(float4) + S_LEN * sizeof(float); // 80 KB
  attn_kernel<<<nblk, 128, shmem, stream>>>(tar_position, current_pos, tar_inp,
                                            ws + WS_G, ws + WS_M);

  tail_kernel<<<(B_N * S_LEN) / 256, 256, 0, stream>>>(current_pos, ws, a4_w, a4_b,
                                                       a5_w, a5_b, outp);
}